// MemoryUpdater_20547123544357
// MI455X (gfx1250) — compile-verified
//
#include <hip/hip_runtime.h>
#include <hip/hip_bf16.h>

typedef __attribute__((ext_vector_type(2))) float v2f;
typedef __attribute__((ext_vector_type(8))) float v8f;

#define Bsz 512
#define Nn  4096
#define L   128
#define KP  388   // 385 padded to multiple of 4 for 16x16x4 f32 WMMA

// ---------------------------------------------------------------- utilities
__global__ void zero_f32(float* __restrict__ p, int n) {
    int i = blockIdx.x * blockDim.x + threadIdx.x;
    if (i < n) p[i] = 0.0f;
}

// pack (128 x 385) row-major weight into (128 x 388) zero-padded
__global__ void pack_w1(const float* __restrict__ w, float* __restrict__ wp) {
    int idx = blockIdx.x * blockDim.x + threadIdx.x;
    if (idx >= 128 * KP) return;
    int n = idx / KP, k = idx % KP;
    wp[idx] = (k < 385) ? w[n * 385 + k] : 0.0f;
}

// ---------------------------------------------------------------- gather
// src_in[b] = [mem[s] | mem[t] | dt[b,s] | x[b,s] | 0 0 0]   (388)
// tar_in[b] = [mem[t] | mem[s] | dt[b,t] | x[b,t] | 0 0 0]
__global__ void gather_pack(const float* __restrict__ x,
                            const float* __restrict__ memory,
                            const float* __restrict__ delta_t,
                            const long long* __restrict__ source,
                            const long long* __restrict__ target,
                            float* __restrict__ src_in,
                            float* __restrict__ tar_in) {
    int b = blockIdx.x;
    int j = threadIdx.x;            // 0..127
    long long s = source[b], t = target[b];
    float sm = memory[(size_t)s * L + j];
    float tm = memory[(size_t)t * L + j];
    float sdt = delta_t[((size_t)b * Nn + (size_t)s) * L + j];
    float tdt = delta_t[((size_t)b * Nn + (size_t)t) * L + j];
    float* so = src_in + (size_t)b * KP;
    float* to = tar_in + (size_t)b * KP;
    so[j] = sm;  so[L + j] = tm;  so[2 * L + j] = sdt;
    to[j] = tm;  to[L + j] = sm;  to[2 * L + j] = tdt;
    if (j == 0) {
        so[384] = x[(size_t)b * Nn + (size_t)s];
        to[384] = x[(size_t)b * Nn + (size_t)t];
        so[385] = 0.f; so[386] = 0.f; so[387] = 0.f;
        to[385] = 0.f; to[386] = 0.f; to[387] = 0.f;
    }
}

// ---------------------------------------------------------------- WMMA GEMM
// C[M x Ntot] = act(A[M x K] @ W[Ntot x K]^T + bias).
// One wave computes a 16x32 tile (two 16x16 accumulators) so each A fragment
// feeds two V_WMMA_F32_16X16X4_F32 issues (2x matrix-op density per load).
// Fragment layout (ISA 7.12.2): A lane l holds A[l&15][2*(l>>4)+v];
// B lane l holds B[2*(l>>4)+v][l&15] = W[l&15][2*(l>>4)+v] for B = W^T.
__global__ void gemm_wmma_f32(const float* __restrict__ A, int lda,
                              const float* __restrict__ W, int ldw,
                              const float* __restrict__ bias,
                              float* __restrict__ C, int ldc,
                              int Mtiles, int Npairs, int K, int relu) {
    int wave = threadIdx.x >> 5;
    int lane = threadIdx.x & 31;
    int tile = blockIdx.x * (blockDim.x >> 5) + wave;
    if (tile >= Mtiles * Npairs) return;          // wave-uniform guard
    int mt = tile / Npairs;
    int np = tile - mt * Npairs;
    int mn = lane & 15;
    int kh = lane >> 4;
    int col0 = np * 32;

    const float* arow  = A + (size_t)(mt * 16 + mn) * lda + 2 * kh;
    const float* wrow0 = W + (size_t)(col0 + mn) * ldw + 2 * kh;
    const float* wrow1 = W + (size_t)(col0 + 16 + mn) * ldw + 2 * kh;

    v8f acc0 = {};
    v8f acc1 = {};
#pragma unroll 4
    for (int k = 0; k < K; k += 4) {
        // CDNA5 prefetch hint (global_prefetch_b8) for the next cache lines
        __builtin_prefetch(arow + k + 32, 0, 1);
        __builtin_prefetch(wrow0 + k + 32, 0, 1);
        __builtin_prefetch(wrow1 + k + 32, 0, 1);
        v2f a  = *(const v2f*)(arow + k);
        v2f b0 = *(const v2f*)(wrow0 + k);
        v2f b1 = *(const v2f*)(wrow1 + k);
        acc0 = __builtin_amdgcn_wmma_f32_16x16x4_f32(
            false, a, false, b0, (short)0, acc0, false, false);
        acc1 = __builtin_amdgcn_wmma_f32_16x16x4_f32(
            false, a, false, b1, (short)0, acc1, false, false);
    }

    int colA = col0 + mn;
    int colB = col0 + 16 + mn;
    float bvA = bias[colA];
    float bvB = bias[colB];
#pragma unroll
    for (int v = 0; v < 8; ++v) {
        int row = mt * 16 + v + 8 * kh;
        float a0 = acc0[v] + bvA;
        float a1 = acc1[v] + bvB;
        if (relu) { a0 = fmaxf(a0, 0.0f); a1 = fmaxf(a1, 0.0f); }
        C[(size_t)row * ldc + colA] = a0;
        C[(size_t)row * ldc + colB] = a1;
    }
}

// ---------------------------------------------------------------- scatter
__global__ void scatter_msgs(const float* __restrict__ msg_src,
                             const float* __restrict__ msg_tar,
                             const long long* __restrict__ source,
                             const long long* __restrict__ target,
                             float* __restrict__ msg_sum,
                             float* __restrict__ cnt) {
    int b = blockIdx.x;
    int j = threadIdx.x;            // 0..127
    long long s = source[b], t = target[b];
    atomicAdd(&msg_sum[(size_t)s * L + j], msg_src[(size_t)b * L + j]);
    atomicAdd(&msg_sum[(size_t)t * L + j], msg_tar[(size_t)b * L + j]);
    if (j == 0) {
        atomicAdd(&cnt[s], 1.0f);
        atomicAdd(&cnt[t], 1.0f);
    }
}

__global__ void compute_agg(const float* __restrict__ msg_sum,
                            const float* __restrict__ cnt,
                            float* __restrict__ agg, int total) {
    int idx = blockIdx.x * blockDim.x + threadIdx.x;
    if (idx >= total) return;
    int node = idx >> 7;
    agg[idx] = msg_sum[idx] / fmaxf(cnt[node], 1.0f);
}

// ---------------------------------------------------------------- GRU combine
__global__ void gru_combine(const float* __restrict__ gi,
                            const float* __restrict__ gh,
                            const float* __restrict__ memory,
                            const float* __restrict__ cnt,
                            float* __restrict__ out, int total) {
    int idx = blockIdx.x * blockDim.x + threadIdx.x;
    if (idx >= total) return;
    int node = idx >> 7, j = idx & (L - 1);
    size_t base = (size_t)node * (3 * L);
    float ir = gi[base + j], iz = gi[base + L + j], inn = gi[base + 2 * L + j];
    float hr = gh[base + j], hz = gh[base + L + j], hn  = gh[base + 2 * L + j];
    float h = memory[idx];
    float r = 1.0f / (1.0f + expf(-(ir + hr)));
    float z = 1.0f / (1.0f + expf(-(iz + hz)));
    float n = tanhf(inn + r * hn);
    float nv = (1.0f - z) * n + z * h;
    out[idx] = (cnt[node] > 0.0f) ? nv : h;
}

// ---------------------------------------------------------------- launch
extern "C" void kernel_launch(void* const* d_in, const int* in_sizes, int n_in,
                              void* d_out, int out_size, void* d_ws, size_t ws_size,
                              hipStream_t stream) {
    const float* x        = (const float*)d_in[0];   // (512,4096,1)
    const float* memory   = (const float*)d_in[1];   // (4096,128)
    const float* delta_t  = (const float*)d_in[2];   // (512,4096,128)
    const float* src_w1   = (const float*)d_in[3];   // (128,385)
    const float* src_b1   = (const float*)d_in[4];
    const float* src_w2   = (const float*)d_in[5];   // (128,128)
    const float* src_b2   = (const float*)d_in[6];
    const float* tar_w1   = (const float*)d_in[7];
    const float* tar_b1   = (const float*)d_in[8];
    const float* tar_w2   = (const float*)d_in[9];
    const float* tar_b2   = (const float*)d_in[10];
    const float* gru_wih  = (const float*)d_in[11];  // (384,128)
    const float* gru_whh  = (const float*)d_in[12];
    const float* gru_bih  = (const float*)d_in[13];
    const float* gru_bhh  = (const float*)d_in[14];
    const long long* source = (const long long*)d_in[15]; // int64 (512,1)
    const long long* target = (const long long*)d_in[16];
    float* out = (float*)d_out;                      // (4096,128)

    float* ws = (float*)d_ws;
    size_t off = 0;
    float* src_in  = ws + off; off += (size_t)Bsz * KP;
    float* tar_in  = ws + off; off += (size_t)Bsz * KP;
    float* w1p_s   = ws + off; off += (size_t)L * KP;
    float* w1p_t   = ws + off; off += (size_t)L * KP;
    float* h1_s    = ws + off; off += (size_t)Bsz * L;
    float* h1_t    = ws + off; off += (size_t)Bsz * L;
    float* msg_s   = ws + off; off += (size_t)Bsz * L;
    float* msg_t   = ws + off; off += (size_t)Bsz * L;
    float* msg_sum = ws + off; off += (size_t)Nn * L;   // contiguous with cnt
    float* cnt     = ws + off; off += (size_t)Nn;
    float* agg     = ws + off; off += (size_t)Nn * L;
    float* gi      = ws + off; off += (size_t)Nn * 3 * L;
    float* gh      = ws + off; off += (size_t)Nn * 3 * L;

    // 1. zero msg_sum + cnt (contiguous region)
    {
        int n = Nn * L + Nn;
        zero_f32<<<(n + 255) / 256, 256, 0, stream>>>(msg_sum, n);
    }
    // 2. pad W1 weights to K=388
    pack_w1<<<(128 * KP + 255) / 256, 256, 0, stream>>>(src_w1, w1p_s);
    pack_w1<<<(128 * KP + 255) / 256, 256, 0, stream>>>(tar_w1, w1p_t);
    // 3. gather inputs
    gather_pack<<<Bsz, L, 0, stream>>>(x, memory, delta_t, source, target,
                                       src_in, tar_in);
    // 4. MLP layer 1 (+ReLU): (512x388)@(388x128)^T -> h1
    //    32 Mtiles x 4 Npairs = 128 wave-tiles, 4 waves/block -> 32 blocks
    gemm_wmma_f32<<<32, 128, 0, stream>>>(src_in, KP, w1p_s, KP, src_b1,
                                          h1_s, L, 32, 4, KP, 1);
    gemm_wmma_f32<<<32, 128, 0, stream>>>(tar_in, KP, w1p_t, KP, tar_b1,
                                          h1_t, L, 32, 4, KP, 1);
    // 5. MLP layer 2: (512x128)@(128x128)^T -> msgs
    gemm_wmma_f32<<<32, 128, 0, stream>>>(h1_s, L, src_w2, L, src_b2,
                                          msg_s, L, 32, 4, L, 0);
    gemm_wmma_f32<<<32, 128, 0, stream>>>(h1_t, L, tar_w2, L, tar_b2,
                                          msg_t, L, 32, 4, L, 0);
    // 6. scatter-add messages + counts
    scatter_msgs<<<Bsz, L, 0, stream>>>(msg_s, msg_t, source, target,
                                        msg_sum, cnt);
    // 7. mean aggregate
    compute_agg<<<(Nn * L + 255) / 256, 256, 0, stream>>>(msg_sum, cnt, agg,
                                                          Nn * L);
    // 8. GRU gates: gi = agg@wih^T + bih, gh = memory@whh^T + bhh
    //    256 Mtiles x 12 Npairs = 3072 wave-tiles -> 768 blocks
    gemm_wmma_f32<<<768, 128, 0, stream>>>(agg, L, gru_wih, L, gru_bih,
                                           gi, 3 * L, 256, 12, L, 0);
    gemm_wmma_f32<<<768, 128, 0, stream>>>(memory, L, gru_whh, L, gru_bhh,
                                           gh, 3 * L, 256, 12, L, 0);
    // 9. pointwise GRU + seen mask
    gru_combine<<<(Nn * L + 255) / 256, 256, 0, stream>>>(gi, gh, memory, cnt,
                                                          out, Nn * L);
}